// HNN_1520418423320
// MI455X (gfx1250) — compile-verified
//
#include <hip/hip_runtime.h>
#include <math.h>

// ---------------------------------------------------------------------------
// Types for CDNA5 WMMA (wave32). V_WMMA_F32_16X16X4_F32:
//   A: 16x4 f32  -> 2 VGPRs/lane (lane M = lane&15, K = 2*(lane>>4)+reg)
//   B: 4x16 f32  -> 2 VGPRs/lane (lane N = lane&15, K = 2*(lane>>4)+reg)
//   C/D: 16x16 f32 -> 8 VGPRs/lane (M = reg + 8*(lane>>4), N = lane&15)
// ---------------------------------------------------------------------------
typedef __attribute__((ext_vector_type(2))) float v2f;
typedef __attribute__((ext_vector_type(8))) float v8f;

#define MODE_PLAIN 0
#define MODE_RELU  1
#define MODE_DUAL  2   // Y = pre-activation, Ysp = softplus(Y)

// LDS row stride for the staged A tile: 514 % 64 == 2, so the 16 rows read for
// a given column land on distinct LDS banks (2r mod 64 distinct for r=0..15).
#define A_STRIDE 514

static __device__ __forceinline__ float sigm(float x) {
    return 1.0f / (1.0f + expf(-x));
}
static __device__ __forceinline__ float softp(float x) {
    return (x > 20.0f) ? x : log1pf(expf(x));
}

// ---------------------------------------------------------------------------
// WMMA fp32 GEMM:  Y[M,N] = X[M,K] @ op(W) + bias
//   transB=1: W is (N,K) row-major, op(W)=W^T   (PyTorch-style x @ w.T)
//   transB=0: W is (K,N) row-major, op(W)=W     (backward-pass x @ w)
// M = 256 (grid.x = 16 row-tiles). grid.y = N/128, block = 128 threads
// (4 waves). The block's shared 16xK A tile is staged once in LDS; each wave
// owns a 16x32 output strip = two independent accumulator chains (2x matrix
// pipe ILP, A fragment amortized over 2 WMMAs). All lanes always active =>
// EXEC all-ones around WMMA, as required by the ISA.
// ---------------------------------------------------------------------------
__global__ __launch_bounds__(128)
void k_gemm_wmma(const float* __restrict__ X, const float* __restrict__ W,
                 const float* __restrict__ bias,
                 float* __restrict__ Y, float* __restrict__ Ysp,
                 int N, int K, int transB, int mode)
{
    __shared__ float As[16 * A_STRIDE];   // up to 16 x 512 fp32 (padded)

    const int tid  = threadIdx.x;
    const int lane = tid & 31;
    const int wid  = tid >> 5;
    const int m0   = blockIdx.x * 16;

    // Cooperative stage of the block-shared A tile (16 rows x K)
    for (int i = tid; i < 16 * K; i += 128) {
        int r = i / K, c = i - r * K;
        As[r * A_STRIDE + c] = X[(size_t)(m0 + r) * K + c];
    }
    __syncthreads();

    const int lo   = lane & 15;        // M (for A) / N (for B) within tile
    const int hi   = lane >> 4;        // selects K pair
    const int kp   = hi << 1;
    const int n0   = blockIdx.y * 128 + wid * 32;
    const int col0 = n0 + lo;
    const int col1 = col0 + 16;

    const float* __restrict__ ar = As + lo * A_STRIDE;

    v8f acc0 = {};
    v8f acc1 = {};

    if (transB) {
        const float* __restrict__ wr0 = W + (size_t)col0 * K;
        const float* __restrict__ wr1 = W + (size_t)col1 * K;
        for (int k0 = 0; k0 < K; k0 += 4) {
            v2f a, b0, b1;
            a.x  = ar[k0 + kp];       a.y  = ar[k0 + kp + 1];
            b0.x = wr0[k0 + kp];      b0.y = wr0[k0 + kp + 1];
            b1.x = wr1[k0 + kp];      b1.y = wr1[k0 + kp + 1];
            acc0 = __builtin_amdgcn_wmma_f32_16x16x4_f32(
                false, a, false, b0, (short)0, acc0, false, false);
            acc1 = __builtin_amdgcn_wmma_f32_16x16x4_f32(
                false, a, false, b1, (short)0, acc1, false, false);
        }
    } else {
        for (int k0 = 0; k0 < K; k0 += 4) {
            v2f a, b0, b1;
            a.x = ar[k0 + kp];        a.y = ar[k0 + kp + 1];
            const float* __restrict__ wr = W + (size_t)(k0 + kp) * N;
            b0.x = wr[col0];          b0.y = wr[N + col0];
            b1.x = wr[col1];          b1.y = wr[N + col1];
            acc0 = __builtin_amdgcn_wmma_f32_16x16x4_f32(
                false, a, false, b0, (short)0, acc0, false, false);
            acc1 = __builtin_amdgcn_wmma_f32_16x16x4_f32(
                false, a, false, b1, (short)0, acc1, false, false);
        }
    }

    const int   mbase = m0 + (hi << 3);
    const float bv0   = bias ? bias[col0] : 0.0f;
    const float bv1   = bias ? bias[col1] : 0.0f;
#pragma unroll
    for (int j = 0; j < 8; ++j) {
        float y0 = acc0[j] + bv0;
        float y1 = acc1[j] + bv1;
        const size_t row = (size_t)(mbase + j) * N;
        if (mode == MODE_RELU) { y0 = fmaxf(y0, 0.0f); y1 = fmaxf(y1, 0.0f); }
        Y[row + col0] = y0;
        Y[row + col1] = y1;
        if (mode == MODE_DUAL) {
            Ysp[row + col0] = softp(y0);
            Ysp[row + col1] = softp(y1);
        }
    }
}

// ---------------------------------------------------------------------------
// Elementwise kernels
// ---------------------------------------------------------------------------
__global__ void k_reparam(const float* __restrict__ mu,
                          const float* __restrict__ logvar,
                          const float* __restrict__ eps,
                          float* __restrict__ q, float* __restrict__ p,
                          float* __restrict__ qs0, int total)
{
    int idx = blockIdx.x * blockDim.x + threadIdx.x;
    if (idx >= total) return;
    int b = idx >> 9, j = idx & 511;
    float z = mu[idx] + expf(0.5f * logvar[idx]) * eps[idx];
    if (j < 256) { q[b * 256 + j] = z; qs0[b * 256 + j] = z; }
    else         { p[b * 256 + (j - 256)] = z; }
}

__global__ void k_concat(const float* __restrict__ q, const float* __restrict__ p,
                         float* __restrict__ zc, int total)
{
    int idx = blockIdx.x * blockDim.x + threadIdx.x;
    if (idx >= total) return;
    int b = idx >> 9, j = idx & 511;
    zc[idx] = (j < 256) ? q[b * 256 + j] : p[b * 256 + (j - 256)];
}

__global__ void k_sigmul(const float* __restrict__ a, const float* __restrict__ v,
                         float* __restrict__ u, int total)
{
    int idx = blockIdx.x * blockDim.x + threadIdx.x;
    if (idx >= total) return;
    u[idx] = sigm(a[idx]) * v[idx];
}

__global__ void k_sigmul_vec(const float* __restrict__ a4, const float* __restrict__ w5,
                             float* __restrict__ u, int total)
{
    int idx = blockIdx.x * blockDim.x + threadIdx.x;
    if (idx >= total) return;
    u[idx] = sigm(a4[idx]) * w5[idx & 127];
}

// out[b,j] = base[b,j] + alpha * g[b, off + j]  (g is (256,512)); optional qs copy
__global__ void k_update(const float* __restrict__ base, const float* __restrict__ g,
                         float* __restrict__ out, float* __restrict__ qs_dst,
                         float alpha, int off, int total)
{
    int idx = blockIdx.x * blockDim.x + threadIdx.x;
    if (idx >= total) return;
    int b = idx >> 8, j = idx & 255;
    float r = base[idx] + alpha * g[b * 512 + off + j];
    out[idx] = r;
    if (qs_dst) qs_dst[idx] = r;
}

// ---------------------------------------------------------------------------
// Direct conv kernels (fp32). NCHW, k=3 pad=1.
// ---------------------------------------------------------------------------
__global__ void k_conv3x3(const float* __restrict__ in, const float* __restrict__ w,
                          const float* __restrict__ bias, float* __restrict__ out,
                          int N, int Cin, int Cout, int H, int Wd, int relu)
{
    int idx = blockIdx.x * blockDim.x + threadIdx.x;
    int total = N * Cout * H * Wd;
    if (idx >= total) return;
    int x  = idx % Wd;
    int y  = (idx / Wd) % H;
    int co = (idx / (Wd * H)) % Cout;
    int n  = idx / (Wd * H * Cout);
    float s = bias[co];
    for (int ci = 0; ci < Cin; ++ci) {
        const float* ip = in + ((size_t)(n * Cin + ci) * H) * Wd;
        const float* wp = w + ((size_t)(co * Cin + ci) * 9);
        for (int ky = 0; ky < 3; ++ky) {
            int iy = y + ky - 1;
            if (iy < 0 || iy >= H) continue;
            for (int kx = 0; kx < 3; ++kx) {
                int ix = x + kx - 1;
                if (ix < 0 || ix >= Wd) continue;
                s += ip[iy * Wd + ix] * wp[ky * 3 + kx];
            }
        }
    }
    out[idx] = relu ? fmaxf(s, 0.0f) : s;
}

// conv3x3 (pad=1) + 2x2 maxpool + relu fused; out dims (H/2, W/2)
__global__ void k_conv3x3_pool(const float* __restrict__ in, const float* __restrict__ w,
                               const float* __restrict__ bias, float* __restrict__ out,
                               int N, int Cin, int Cout, int H, int Wd)
{
    int Ho = H >> 1, Wo = Wd >> 1;
    int idx = blockIdx.x * blockDim.x + threadIdx.x;
    int total = N * Cout * Ho * Wo;
    if (idx >= total) return;
    int x2 = idx % Wo;
    int y2 = (idx / Wo) % Ho;
    int co = (idx / (Wo * Ho)) % Cout;
    int n  = idx / (Wo * Ho * Cout);
    float m = -3.0e38f;
    for (int py = 0; py < 2; ++py) {
        for (int px = 0; px < 2; ++px) {
            int y = 2 * y2 + py, x = 2 * x2 + px;
            float s = 0.0f;
            for (int ci = 0; ci < Cin; ++ci) {
                const float* ip = in + ((size_t)(n * Cin + ci) * H) * Wd;
                const float* wp = w + ((size_t)(co * Cin + ci) * 9);
                for (int ky = 0; ky < 3; ++ky) {
                    int iy = y + ky - 1;
                    if (iy < 0 || iy >= H) continue;
                    for (int kx = 0; kx < 3; ++kx) {
                        int ix = x + kx - 1;
                        if (ix < 0 || ix >= Wd) continue;
                        s += ip[iy * Wd + ix] * wp[ky * 3 + kx];
                    }
                }
            }
            m = fmaxf(m, s);
        }
    }
    out[idx] = fmaxf(m + bias[co], 0.0f);
}

// ConvTranspose: k=5, stride 2, pad (2,3); weight layout (Cin, Cout, 5, 5).
// Hout = 2*Hin.  y[n,o,yo,xo] = sum w[c,o,4-ky,4-kx] * x[n,c,(yo+ky-2)/2,(xo+kx-2)/2]
__global__ void k_convT5(const float* __restrict__ in, const float* __restrict__ w,
                         const float* __restrict__ bias, float* __restrict__ out,
                         int N, int C, int Hin)
{
    int Hout = Hin * 2;
    int idx = blockIdx.x * blockDim.x + threadIdx.x;
    int total = N * C * Hout * Hout;
    if (idx >= total) return;
    int x  = idx % Hout;
    int y  = (idx / Hout) % Hout;
    int co = (idx / (Hout * Hout)) % C;
    int n  = idx / (Hout * Hout * C);
    float s = bias[co];
    for (int ky = 0; ky < 5; ++ky) {
        int t = y + ky - 2;
        if (t < 0 || (t & 1)) continue;
        int iy = t >> 1;
        if (iy >= Hin) continue;
        for (int kx = 0; kx < 5; ++kx) {
            int u = x + kx - 2;
            if (u < 0 || (u & 1)) continue;
            int ix = u >> 1;
            if (ix >= Hin) continue;
            int widx = ((4 - ky) * 5 + (4 - kx));
            for (int ci = 0; ci < C; ++ci) {
                s += in[((size_t)(n * C + ci) * Hin + iy) * Hin + ix]
                   * w[((size_t)(ci * C + co) * 25) + widx];
            }
        }
    }
    out[idx] = fmaxf(s, 0.0f);
}

// Final decoder conv: (7680,16,32,32) -> sigmoid -> permuted into d_out result
// d row i = f*256 + b ; result[b, f*3 + co, y, x]
__global__ void k_conv_final(const float* __restrict__ in, const float* __restrict__ w,
                             const float* __restrict__ bias, float* __restrict__ outres)
{
    int idx = blockIdx.x * blockDim.x + threadIdx.x;
    const int total = 7680 * 3 * 32 * 32;
    if (idx >= total) return;
    int x   = idx % 32;
    int y   = (idx / 32) % 32;
    int co  = (idx / 1024) % 3;
    int img = idx / (3 * 1024);
    int f   = img / 256;
    int b   = img % 256;
    float s = bias[co];
    for (int ci = 0; ci < 16; ++ci) {
        const float* ip = in + ((size_t)(img * 16 + ci) * 1024);
        const float* wp = w + ((size_t)(co * 16 + ci) * 9);
        for (int ky = 0; ky < 3; ++ky) {
            int iy = y + ky - 1;
            if (iy < 0 || iy >= 32) continue;
            for (int kx = 0; kx < 3; ++kx) {
                int ix = x + kx - 1;
                if (ix < 0 || ix >= 32) continue;
                s += ip[iy * 32 + ix] * wp[ky * 3 + kx];
            }
        }
    }
    outres[(((size_t)b * 90 + f * 3 + co) * 32 + y) * 32 + x] = sigm(s);
}

// ---------------------------------------------------------------------------
// Host orchestration
// ---------------------------------------------------------------------------
extern "C" void kernel_launch(void* const* d_in, const int* in_sizes, int n_in,
                              void* d_out, int out_size, void* d_ws, size_t ws_size,
                              hipStream_t stream)
{
    (void)in_sizes; (void)n_in; (void)out_size; (void)ws_size;

    const float* X    = (const float*)d_in[0];
    const float* EPS  = (const float*)d_in[1];
    const float* ew[5] = {(const float*)d_in[2],(const float*)d_in[4],(const float*)d_in[6],
                          (const float*)d_in[8],(const float*)d_in[10]};
    const float* eb[5] = {(const float*)d_in[3],(const float*)d_in[5],(const float*)d_in[7],
                          (const float*)d_in[9],(const float*)d_in[11]};
    const float* f1w1=(const float*)d_in[12]; const float* f1b1=(const float*)d_in[13];
    const float* f1w2=(const float*)d_in[14]; const float* f1b2=(const float*)d_in[15];
    const float* f2w1=(const float*)d_in[16]; const float* f2b1=(const float*)d_in[17];
    const float* f2w2=(const float*)d_in[18]; const float* f2b2=(const float*)d_in[19];
    const float* hw1=(const float*)d_in[20]; const float* hb1=(const float*)d_in[21];
    const float* hw2=(const float*)d_in[22]; const float* hb2=(const float*)d_in[23];
    const float* hw3=(const float*)d_in[24]; const float* hb3=(const float*)d_in[25];
    const float* hw4=(const float*)d_in[26]; const float* hb4=(const float*)d_in[27];
    const float* hw5=(const float*)d_in[28];
    const float* dw[6] = {(const float*)d_in[30],(const float*)d_in[32],(const float*)d_in[34],
                          (const float*)d_in[36],(const float*)d_in[38],(const float*)d_in[40]};
    const float* db[6] = {(const float*)d_in[31],(const float*)d_in[33],(const float*)d_in[35],
                          (const float*)d_in[37],(const float*)d_in[39],(const float*)d_in[41]};

    float* RESULT = (float*)d_out;                 // 256*90*32*32
    float* MU     = RESULT + 23592960;             // 256*512
    float* LOGVAR = MU + 131072;                   // 256*512

    // ---- workspace layout (floats) ----
    float* ws = (float*)d_ws;
    float* Hb  = ws;            // (256,512) encoder features
    float* T1  = Hb  + 131072;  // (256,256) fc hidden
    float* Qa  = T1  + 65536;   // (256,256)
    float* Pa  = Qa  + 65536;
    float* Qb  = Pa  + 65536;
    float* Pb  = Qb  + 65536;
    float* ZC  = Pb  + 65536;   // (256,512)
    float* A1  = ZC  + 131072;  float* S1 = A1 + 131072;   // (256,512)
    float* A2  = S1  + 131072;  float* S2 = A2 + 65536;    // (256,256)
    float* A3  = S2  + 65536;   float* S3 = A3 + 32768;    // (256,128)
    float* A4  = S3  + 32768;                               // (256,128)
    float* Ub  = A4  + 32768;   // (256,512) ping
    float* Vb  = Ub  + 131072;  // (256,512) pong
    float* G   = Vb  + 131072;  // (256,512) hamiltonian gradient
    float* QS  = G   + 131072;  // (30,256,256) frames
    float* AR  = QS  + 1966080; // big arena: encoder then decoder temporaries

    // encoder temporaries (in arena)
    float* E1 = AR;              // (256,64,32,32)
    float* E2 = E1 + 16777216;   // (256,64,32,32)
    float* E3 = E2 + 16777216;   // (256,64,16,16)
    float* E4 = E3 + 4194304;    // (256,64,8,8)
    // decoder temporaries (reuse arena after HNN)
    float* D1 = AR;              // (7680,16,8,8)
    float* D2 = D1 + 7864320;
    float* D3 = D2 + 7864320;    // (7680,16,16,16)
    float* D4 = D3 + 31457280;
    float* D5 = D4 + 31457280;   // (7680,16,32,32)

    auto nb = [](int n){ return (n + 255) / 256; };
    auto gemm = [&](const float* Xp, const float* Wp, const float* Bp,
                    float* Yp, float* Yspp, int N, int K, int transB, int mode){
        dim3 grid(16, N / 128);
        k_gemm_wmma<<<grid, 128, 0, stream>>>(Xp, Wp, Bp, Yp, Yspp, N, K, transB, mode);
    };

    // ---------------- encoder ----------------
    k_conv3x3<<<nb(256*64*1024),256,0,stream>>>(X,  ew[0], eb[0], E1, 256, 90, 64, 32, 32, 1);
    k_conv3x3<<<nb(256*64*1024),256,0,stream>>>(E1, ew[1], eb[1], E2, 256, 64, 64, 32, 32, 1);
    k_conv3x3_pool<<<nb(256*64*256),256,0,stream>>>(E2, ew[2], eb[2], E3, 256, 64, 64, 32, 32);
    k_conv3x3_pool<<<nb(256*64*64), 256,0,stream>>>(E3, ew[3], eb[3], E4, 256, 64, 64, 16, 16);
    k_conv3x3_pool<<<nb(256*32*16), 256,0,stream>>>(E4, ew[4], eb[4], Hb, 256, 64, 32,  8,  8);

    // ---------------- FC heads (WMMA) ----------------
    gemm(Hb, f1w1, f1b1, T1, nullptr, 256, 512, 1, MODE_RELU);
    gemm(T1, f1w2, f1b2, MU, nullptr, 512, 256, 1, MODE_PLAIN);
    gemm(Hb, f2w1, f2b1, T1, nullptr, 256, 512, 1, MODE_RELU);
    gemm(T1, f2w2, f2b2, LOGVAR, nullptr, 512, 256, 1, MODE_PLAIN);

    k_reparam<<<nb(131072),256,0,stream>>>(MU, LOGVAR, EPS, Qa, Pa, QS, 131072);

    // ---------------- Hamiltonian gradient (8 WMMA GEMMs) ----------------
    auto ham_grad = [&](const float* zc){
        gemm(zc, hw1, hb1, A1, S1, 512, 512, 1, MODE_DUAL);
        gemm(S1, hw2, hb2, A2, S2, 256, 512, 1, MODE_DUAL);
        gemm(S2, hw3, hb3, A3, S3, 128, 256, 1, MODE_DUAL);
        gemm(S3, hw4, hb4, A4, nullptr, 128, 128, 1, MODE_PLAIN);
        k_sigmul_vec<<<nb(32768),256,0,stream>>>(A4, hw5, Ub, 32768);
        gemm(Ub, hw4, nullptr, Vb, nullptr, 128, 128, 0, MODE_PLAIN);
        k_sigmul<<<nb(32768),256,0,stream>>>(A3, Vb, Ub, 32768);
        gemm(Ub, hw3, nullptr, Vb, nullptr, 256, 128, 0, MODE_PLAIN);
        k_sigmul<<<nb(65536),256,0,stream>>>(A2, Vb, Ub, 65536);
        gemm(Ub, hw2, nullptr, Vb, nullptr, 512, 256, 0, MODE_PLAIN);
        k_sigmul<<<nb(131072),256,0,stream>>>(A1, Vb, Ub, 131072);
        gemm(Ub, hw1, nullptr, G, nullptr, 512, 512, 0, MODE_PLAIN);
    };

    const float DT = 0.125f;
    // ---------------- leapfrog integration, 29 steps ----------------
    for (int i = 1; i < 30; ++i) {
        k_concat<<<nb(131072),256,0,stream>>>(Qa, Pa, ZC, 131072);
        ham_grad(ZC);
        k_update<<<nb(65536),256,0,stream>>>(Pa, G, Pb, nullptr, -0.5f * DT, 256, 65536);

        k_concat<<<nb(131072),256,0,stream>>>(Qa, Pb, ZC, 131072);
        ham_grad(ZC);
        k_update<<<nb(65536),256,0,stream>>>(Qa, G, Qb, QS + (size_t)i * 65536, DT, 0, 65536);

        k_concat<<<nb(131072),256,0,stream>>>(Qb, Pb, ZC, 131072);
        ham_grad(ZC);
        k_update<<<nb(65536),256,0,stream>>>(Pb, G, Pa, nullptr, -0.5f * DT, 256, 65536);

        float* t = Qa; Qa = Qb; Qb = t;   // Qa = q_i, Pa = p_i
    }

    // ---------------- decoder (QS viewed as (7680,16,4,4)) ----------------
    k_convT5<<<nb(7680*16*64),  256,0,stream>>>(QS, dw[0], db[0], D1, 7680, 16, 4);
    k_conv3x3<<<nb(7680*16*64), 256,0,stream>>>(D1, dw[1], db[1], D2, 7680, 16, 16, 8, 8, 1);
    k_convT5<<<nb(7680*16*256), 256,0,stream>>>(D2, dw[2], db[2], D3, 7680, 16, 8);
    k_conv3x3<<<nb(7680*16*256),256,0,stream>>>(D3, dw[3], db[3], D4, 7680, 16, 16, 16, 16, 1);
    k_convT5<<<nb(7680*16*1024),256,0,stream>>>(D4, dw[4], db[4], D5, 7680, 16, 16);
    k_conv_final<<<nb(7680*3*1024),256,0,stream>>>(D5, dw[5], db[5], RESULT);
}